// EdgeReadout_67808943669265
// MI455X (gfx1250) — compile-verified
//
#include <hip/hip_runtime.h>

// CDNA5 / gfx1250, wave32 only.
typedef float v2f __attribute__((ext_vector_type(2)));
typedef float v8f __attribute__((ext_vector_type(8)));

// Only check the builtin on the DEVICE pass; host pass has no AMDGCN builtins.
#if defined(__HIP_DEVICE_COMPILE__)
#if !__has_builtin(__builtin_amdgcn_wmma_f32_16x16x4_f32)
#error "device pass: missing __builtin_amdgcn_wmma_f32_16x16x4_f32"
#endif
#endif

// Edge readout MLP:
//   pair = concat(x[src], x[dst])            [E, 64]
//   h    = relu(pair @ W1 + b1)              [E, 32]
//   out  = h @ W2 + b2                       [E, 2]
//
// One wave handles 16 edges per tile:
//   A (16x64 pair matrix) fed as 16 K-slices of 16x4 f32,
//   B (W1, 64x32) pre-staged in registers as 16x2 fragments of 4x16,
//   C (16x16 f32 x 2 N-tiles) accumulated via V_WMMA_F32_16X16X4_F32,
//   second layer (32->2) via VALU + xor-butterfly cross-lane reduction.
__global__ __launch_bounds__(256) void edge_mlp_wmma_kernel(
    const float* __restrict__ x,          // [N_NODES, 32]
    const long long* __restrict__ ei,     // [2, E] int64
    const float* __restrict__ W1,         // [64, 32] row-major
    const float* __restrict__ b1,         // [32]
    const float* __restrict__ W2,         // [32, 2] row-major
    const float* __restrict__ b2,         // [2]
    float* __restrict__ out,              // [E, 2]
    int n_edges)
{
#if defined(__HIP_DEVICE_COMPILE__)
    const int lane = threadIdx.x & 31;
    const int half = lane >> 4;           // 0: lanes 0-15, 1: lanes 16-31
    const int lm   = lane & 15;

    // ---- Stage B fragments (W1) in registers -----------------------------
    // 16x16x4 f32 B layout (4x16, KxN), 2 VGPRs:
    //   v0: lanes 0-15 -> K=0, lanes 16-31 -> K=2
    //   v1: lanes 0-15 -> K=1, lanes 16-31 -> K=3
    v2f bf[16][2];
#pragma unroll
    for (int k = 0; k < 16; ++k) {
        const int r0 = 4 * k + 2 * half;
#pragma unroll
        for (int n = 0; n < 2; ++n) {
            const int c = 16 * n + lm;
            bf[k][n].x = W1[r0 * 32 + c];
            bf[k][n].y = W1[(r0 + 1) * 32 + c];
        }
    }
    const float b1v0 = b1[lm];            // bias for N-tile 0 column lm
    const float b1v1 = b1[16 + lm];       // bias for N-tile 1 column 16+lm
    const float w2_lo0 = W2[lm * 2 + 0];  // W2[c=lm][0]
    const float w2_lo1 = W2[lm * 2 + 1];
    const float w2_hi0 = W2[(16 + lm) * 2 + 0];
    const float w2_hi1 = W2[(16 + lm) * 2 + 1];
    const float b2v0 = b2[0];
    const float b2v1 = b2[1];

    const int n_tiles = (n_edges + 15) >> 4;
    const int wave_id = (int)((blockIdx.x * blockDim.x + threadIdx.x) >> 5);
    const int n_waves = (int)((gridDim.x * blockDim.x) >> 5);

    for (int t = wave_id; t < n_tiles; t += n_waves) {
        const int e0 = t << 4;

        // Each lane owns pair-matrix row M = lane&15 (A layout: both halves
        // cover M=0..15; halves differ only in which K pair they hold).
        int eM = e0 + lm;
        if (eM >= n_edges) eM = n_edges - 1;      // clamp tail (loads only)
        const long long s = ei[eM];
        const long long d = ei[(long long)n_edges + eM];
        const float* xs = x + s * 32 + 2 * half;  // +0/+2 K-pair offset
        const float* xd = x + d * 32 + 2 * half;

        // C init = b1 broadcast (bias folded into accumulator).
        v8f acc0, acc1;
#pragma unroll
        for (int i = 0; i < 8; ++i) { acc0[i] = b1v0; acc1[i] = b1v1; }

        // K loop: 16 slices of K=4. k<8 reads x[src] cols 4k..; k>=8 reads
        // x[dst]. Each lane does one aligned float2 (b64) gather per slice.
#pragma unroll
        for (int k = 0; k < 16; ++k) {
            const float* p = (k < 8) ? (xs + 4 * k) : (xd + 4 * (k - 8));
            v2f a = *reinterpret_cast<const v2f*>(p);
            acc0 = __builtin_amdgcn_wmma_f32_16x16x4_f32(
                false, a, false, bf[k][0], (short)0, acc0, false, false);
            acc1 = __builtin_amdgcn_wmma_f32_16x16x4_f32(
                false, a, false, bf[k][1], (short)0, acc1, false, false);
        }

        // ---- ReLU + second layer (32 -> 2) + cross-lane reduction --------
        // C layout: VGPR i, lanes 0-15 -> row i, lanes 16-31 -> row i+8;
        // lane holds h[row][lm] (tile0) and h[row][16+lm] (tile1).
        float r0 = 0.0f, r1 = 0.0f;
#pragma unroll
        for (int i = 0; i < 8; ++i) {
            const float h0 = fmaxf(acc0[i], 0.0f);
            const float h1 = fmaxf(acc1[i], 0.0f);
            float p0 = h0 * w2_lo0 + h1 * w2_hi0;
            float p1 = h0 * w2_lo1 + h1 * w2_hi1;
            // Butterfly sum over the 16 lanes of each half (masks < 16 keep
            // the two halves independent).
#pragma unroll
            for (int m = 8; m >= 1; m >>= 1) {
                p0 += __shfl_xor(p0, m, 32);
                p1 += __shfl_xor(p1, m, 32);
            }
            if (lm == i) { r0 = p0; r1 = p1; }   // park row i+8*half
        }

        // Lanes 0-7 hold rows 0-7, lanes 16-23 hold rows 8-15.
        const int row = half * 8 + lm;
        const int e = e0 + row;
        if (lm < 8 && e < n_edges) {
            float2 o;
            o.x = r0 + b2v0;
            o.y = r1 + b2v1;
            *reinterpret_cast<float2*>(out + 2 * (size_t)e) = o;
        }
    }
#endif // __HIP_DEVICE_COMPILE__
}

extern "C" void kernel_launch(void* const* d_in, const int* in_sizes, int n_in,
                              void* d_out, int out_size, void* d_ws, size_t ws_size,
                              hipStream_t stream) {
    const float*     x  = (const float*)d_in[0];
    const long long* ei = (const long long*)d_in[1];   // int64 edge_index [2,E]
    const float*     W1 = (const float*)d_in[2];
    const float*     b1 = (const float*)d_in[3];
    const float*     W2 = (const float*)d_in[4];
    const float*     b2 = (const float*)d_in[5];
    float* out = (float*)d_out;

    const int n_edges = in_sizes[1] / 2;

    // Gather-latency bound: launch plenty of waves (8192) so the L2 gather
    // pipeline stays full; each wave grid-strides over 16-edge tiles.
    const int threads = 256;   // 8 waves per workgroup (wave32)
    const int blocks  = 1024;
    hipLaunchKernelGGL(edge_mlp_wmma_kernel, dim3(blocks), dim3(threads), 0, stream,
                       x, ei, W1, b1, W2, b2, out, n_edges);
}